// Rwkv6SelfAttention_60644938220063
// MI455X (gfx1250) — compile-verified
//
#include <hip/hip_runtime.h>
#include <hip/hip_bf16.h>

// RWKV6 single-token forward for gfx1250 (MI455X).
// Roofline: ~350 MB of fp32 weight traffic vs ~0.17 GFLOP -> pure HBM-bandwidth
// problem (~15us at 23.3 TB/s). Heavy GEMVs stream weights once, coalesced,
// accumulating with V_WMMA_F32_16X16X4_F32 (M-redundancy is free). The four
// independent 64MB GEMVs are fused into ONE launch with an 8-way K-split
// (8192 waves) so a single kernel can saturate the memory system; partials are
// reduced deterministically (fixed order, no float atomics).

#define H  4096
#define HS 64
#define NH 64
#define TM 64
#define TD 128
#define KC 512          // K-chunk per wave (compile-time: no tail branches)
#define NCHUNK (H / KC) // 8

typedef __attribute__((ext_vector_type(2))) float v2f;
typedef __attribute__((ext_vector_type(8))) float v8f;

// ---------------------------------------------------------------------------
// Kernel 0: LayerNorm + token-shift prep.
// ---------------------------------------------------------------------------
__global__ __launch_bounds__(256) void ln_kernel(
    const float* __restrict__ x, const float* __restrict__ s1,
    const float* __restrict__ lw, const float* __restrict__ lb,
    const float* __restrict__ tmx,
    float* __restrict__ xn, float* __restrict__ sx, float* __restrict__ xxx,
    float* __restrict__ s1out)
{
    __shared__ float red[256];
    const int t = threadIdx.x;

    float s = 0.f;
    for (int i = t; i < H; i += 256) s += x[i];
    red[t] = s; __syncthreads();
    for (int st = 128; st > 0; st >>= 1) {
        if (t < st) red[t] += red[t + st];
        __syncthreads();
    }
    const float mean = red[0] * (1.f / H);
    __syncthreads();

    s = 0.f;
    for (int i = t; i < H; i += 256) { float d = x[i] - mean; s += d * d; }
    red[t] = s; __syncthreads();
    for (int st = 128; st > 0; st >>= 1) {
        if (t < st) red[t] += red[t + st];
        __syncthreads();
    }
    const float rstd = rsqrtf(red[0] * (1.f / H) + 1e-5f);

    for (int i = t; i < H; i += 256) {
        float n = (x[i] - mean) * rstd * lw[i] + lb[i];
        xn[i] = n;
        s1out[i] = n;
        float d = s1[i] - n;
        sx[i] = d;
        xxx[i] = n + d * tmx[i];
    }
}

// ---------------------------------------------------------------------------
// WMMA GEMV tile body: partial y[n0..n0+15] over one K-chunk of 512.
// A(16x4) = broadcast x chunk (rows identical), B(4x16) = natural W tile.
// Row 0 of D (c[0], lanes 0-15) carries the partial dot products.
// ---------------------------------------------------------------------------
__device__ __forceinline__ void gemv_tile(
    const float* __restrict__ W, const float* __restrict__ x,
    float* __restrict__ part, int N, int kbase, float* xs)
{
    const int tid = threadIdx.y * 32 + threadIdx.x;
    for (int i = tid; i < KC; i += 256) xs[i] = x[kbase + i];
    __syncthreads();

    const int wave = blockIdx.x * 8 + threadIdx.y;
    const int ntile = N >> 4;
    int n0 = wave * 16;
    if (wave >= ntile) n0 = (ntile - 1) * 16;  // extra waves repeat last tile
    const int lane = threadIdx.x;
    const int half = lane >> 4;                 // 0: K rows {0,1}; 1: {2,3}
    const int col  = lane & 15;

    const float* wp = W + (size_t)(kbase + 2 * half) * N + n0 + col;
    v8f c = {0.f, 0.f, 0.f, 0.f, 0.f, 0.f, 0.f, 0.f};

#pragma unroll 4
    for (int i0 = 0; i0 < KC; i0 += 4) {
        v2f a, b;
        a.x = xs[i0 + 2 * half];               // A[m][2h]
        a.y = xs[i0 + 2 * half + 1];           // A[m][2h+1]
        b.x = wp[0];                           // B[2h][col]
        b.y = wp[N];                           // B[2h+1][col]
        c = __builtin_amdgcn_wmma_f32_16x16x4_f32(
                false, a, false, b, (short)0, c, false, false);
        wp += (size_t)4 * N;
    }

    if (lane < 16) part[n0 + lane] = c[0];
}

// Single-matrix K-split GEMV: grid = (ceil(N/16/8), NCHUNK)
__global__ __launch_bounds__(256) void wmma_gemv_part(
    const float* __restrict__ W, const float* __restrict__ x,
    float* __restrict__ part, int N)
{
    __shared__ float xs[KC];
    gemv_tile(W, x, part + (size_t)blockIdx.y * N, N, blockIdx.y * KC, xs);
}

// Fused 4-matrix K-split GEMV: grid = (N/16/8, NCHUNK, 4); part[mat][chunk][N]
__global__ __launch_bounds__(256) void wmma_gemv_part4(
    const float* __restrict__ W0, const float* __restrict__ W1,
    const float* __restrict__ W2, const float* __restrict__ W3,
    const float* __restrict__ x0, const float* __restrict__ x1,
    const float* __restrict__ x2, const float* __restrict__ x3,
    float* __restrict__ part, int N)
{
    __shared__ float xs[KC];
    const int mat = blockIdx.z;
    const float* W = (mat == 0) ? W0 : (mat == 1) ? W1 : (mat == 2) ? W2 : W3;
    const float* x = (mat == 0) ? x0 : (mat == 1) ? x1 : (mat == 2) ? x2 : x3;
    gemv_tile(W, x, part + ((size_t)mat * NCHUNK + blockIdx.y) * N,
              N, blockIdx.y * KC, xs);
}

// Deterministic chunk reduction: y[n] = act(sum_c part[c][n]) (+residual)
// grid.y selects independent (part, y) pairs with strides nchunks*N / N.
__global__ __launch_bounds__(256) void reduce_kernel(
    const float* __restrict__ part, float* __restrict__ y, int N, int nchunks,
    int act, const float* __restrict__ residual)
{
    const int n = blockIdx.x * 256 + threadIdx.x;
    if (n >= N) return;
    part += (size_t)blockIdx.y * nchunks * N;
    y    += (size_t)blockIdx.y * N;
    float s = 0.f;
    for (int c = 0; c < nchunks; ++c) s += part[(size_t)c * N + n];
    if (act == 1) s = tanhf(s);
    if (residual) s += residual[n];
    y[n] = s;
}

// ---------------------------------------------------------------------------
// Kernel 2: m_s = xn + sx*( mix[s]·W_maa2[s] + maa_s ),  s = 0..4 (w,k,v,r,g)
// ---------------------------------------------------------------------------
__global__ __launch_bounds__(256) void mixmaa_kernel(
    const float* __restrict__ xn, const float* __restrict__ sx,
    const float* __restrict__ mix, const float* __restrict__ Wmaa2,
    const float* __restrict__ maa0, const float* __restrict__ maa1,
    const float* __restrict__ maa2, const float* __restrict__ maa3,
    const float* __restrict__ maa4,
    float* __restrict__ m)
{
    const int s = blockIdx.y;
    const int j = blockIdx.x * 256 + threadIdx.x;
    __shared__ float ms[TM];
    if (threadIdx.x < TM) ms[threadIdx.x] = mix[s * TM + threadIdx.x];
    __syncthreads();

    const float* wp = Wmaa2 + (size_t)s * TM * H + j;
    float acc = 0.f;
#pragma unroll 8
    for (int e = 0; e < TM; ++e) acc += ms[e] * wp[(size_t)e * H];

    const float* maa = (s == 0) ? maa0 : (s == 1) ? maa1 : (s == 2) ? maa2
                      : (s == 3) ? maa3 : maa4;
    m[(size_t)s * H + j] = xn[j] + sx[j] * (acc + maa[j]);
}

// ---------------------------------------------------------------------------
// Kernel 4: per-head decay, state update, attention readout, InstanceNorm,
// SiLU gate. One block per head, 64 threads.
// ---------------------------------------------------------------------------
__global__ __launch_bounds__(64) void head_kernel(
    const float* __restrict__ r, const float* __restrict__ k,
    const float* __restrict__ v, const float* __restrict__ gpre,
    const float* __restrict__ dd1, const float* __restrict__ Wd2,
    const float* __restrict__ tdec, const float* __restrict__ faaaa,
    const float* __restrict__ state2,
    const float* __restrict__ lnxw, const float* __restrict__ lnxb,
    float* __restrict__ ox, float* __restrict__ s2out)
{
    __shared__ float dd1s[TD];
    __shared__ float td[HS], rs[HS], ks[HS], red[HS];
    const int h = blockIdx.x;
    const int t = threadIdx.x;

    dd1s[t]      = dd1[t];
    dd1s[t + 64] = dd1[t + 64];
    rs[t] = r[h * HS + t];
    ks[t] = k[h * HS + t];
    __syncthreads();

    // dd = dd1 · Wd2[:, h*64+t];  td = exp(-exp(time_decay + dd))
    float dd = 0.f;
    const float* wcol = Wd2 + h * HS + t;
#pragma unroll 8
    for (int e = 0; e < TD; ++e) dd += dd1s[e] * wcol[(size_t)e * H];
    td[t] = expf(-expf(tdec[h * HS + t] + dd));

    // rkf = sum_i r*k*faaaa
    red[t] = rs[t] * ks[t] * faaaa[h * HS + t];
    __syncthreads();
    for (int st = 32; st > 0; st >>= 1) {
        if (t < st) red[t] += red[t + st];
        __syncthreads();
    }
    const float rkf = red[0];
    __syncthreads();

    // thread t == output dim d
    const float vd = v[h * HS + t];
    float acc = 0.f;
    const float* s2p = state2 + (size_t)h * HS * HS + t;
    float*       s2o = s2out  + (size_t)h * HS * HS + t;
#pragma unroll 4
    for (int i = 0; i < HS; ++i) {
        float s2 = s2p[(size_t)i * HS];
        acc += rs[i] * s2;
        s2o[(size_t)i * HS] = ks[i] * vd + s2 * td[i];
    }
    const float outd = rkf * vd + acc;

    // InstanceNorm over the 64 dims of this head
    red[t] = outd; __syncthreads();
    for (int st = 32; st > 0; st >>= 1) {
        if (t < st) red[t] += red[t + st];
        __syncthreads();
    }
    const float mean = red[0] * (1.f / HS);
    __syncthreads();
    float d0 = outd - mean;
    red[t] = d0 * d0; __syncthreads();
    for (int st = 32; st > 0; st >>= 1) {
        if (t < st) red[t] += red[t + st];
        __syncthreads();
    }
    const float rstd = rsqrtf(red[0] * (1.f / HS) + 1e-5f);

    const float on = d0 * rstd * lnxw[h * HS + t] + lnxb[h * HS + t];
    const float z  = gpre[h * HS + t];
    const float silu = z / (1.f + expf(-z));
    ox[h * HS + t] = on * silu;
}

// ---------------------------------------------------------------------------
extern "C" void kernel_launch(void* const* d_in, const int* in_sizes, int n_in,
                              void* d_out, int out_size, void* d_ws, size_t ws_size,
                              hipStream_t stream)
{
    const float* x     = (const float*)d_in[0];
    const float* s1    = (const float*)d_in[1];
    const float* st2   = (const float*)d_in[2];
    const float* ln1w  = (const float*)d_in[3];
    const float* ln1b  = (const float*)d_in[4];
    const float* tmx   = (const float*)d_in[5];
    const float* tmw   = (const float*)d_in[6];
    const float* tmk   = (const float*)d_in[7];
    const float* tmv   = (const float*)d_in[8];
    const float* tmr   = (const float*)d_in[9];
    const float* tmg   = (const float*)d_in[10];
    const float* Wmaa1 = (const float*)d_in[11];
    const float* Wmaa2 = (const float*)d_in[12];
    const float* tdec  = (const float*)d_in[13];
    const float* Wd1   = (const float*)d_in[14];
    const float* Wd2   = (const float*)d_in[15];
    const float* faaaa = (const float*)d_in[16];
    const float* Wr    = (const float*)d_in[17];
    const float* Wk    = (const float*)d_in[18];
    const float* Wv    = (const float*)d_in[19];
    const float* Wg    = (const float*)d_in[20];
    const float* Wo    = (const float*)d_in[21];
    const float* lnxw  = (const float*)d_in[22];
    const float* lnxb  = (const float*)d_in[23];

    float* out_x  = (float*)d_out;            // x + o            (4096)
    float* out_s1 = out_x + H;                // state1_out       (4096)
    float* out_s2 = out_s1 + H;               // state2_out       (64*64*64)

    float* w        = (float*)d_ws;
    float* xn       = w;                      // 4096
    float* sx       = xn + H;                 // 4096
    float* xxx      = sx + H;                 // 4096
    float* mix      = xxx + H;                // 320  (+pad)
    float* m        = mix + 512;              // 5*4096  (mw,mk,mv,mr,mg)
    float* dd1      = m + 5 * H;              // 128  (+pad)
    float* rr       = dd1 + 256;              // 4096  | rr,kk,vv,gp contiguous
    float* kk       = rr + H;                 // 4096  | (reduce_kernel grid.y)
    float* vv       = kk + H;                 // 4096
    float* gp       = vv + H;                 // 4096
    float* ox       = gp + H;                 // 4096
    float* part_mix = ox + H;                 // NCHUNK*320
    float* part_dd  = part_mix + NCHUNK * 320;// NCHUNK*128
    float* part_big = part_dd + NCHUNK * 128; // 4*NCHUNK*4096
    float* part_o   = part_big + 4 * NCHUNK * H; // NCHUNK*4096

    const dim3 wblk(32, 8);

    // 0) LayerNorm + shifts
    ln_kernel<<<1, 256, 0, stream>>>(x, s1, ln1w, ln1b, tmx, xn, sx, xxx, out_s1);

    // 1) mix = tanh(xxx @ W_maa1)   K=4096, N=320
    wmma_gemv_part<<<dim3(3, NCHUNK), wblk, 0, stream>>>(Wmaa1, xxx, part_mix, 5 * TM);
    reduce_kernel<<<dim3(2, 1), 256, 0, stream>>>(part_mix, mix, 5 * TM, NCHUNK, 1, nullptr);

    // 2) five mixed vectors m[s]
    mixmaa_kernel<<<dim3(H / 256, 5), 256, 0, stream>>>(
        xn, sx, mix, Wmaa2, tmw, tmk, tmv, tmr, tmg, m);

    // 3) fused heavy GEMVs: r,k,v,g_pre (4 x 64MB, 8192 waves in one launch)
    wmma_gemv_part4<<<dim3(H / 128, NCHUNK, 4), wblk, 0, stream>>>(
        Wr, Wk, Wv, Wg,
        m + 3 * H /*mr*/, m + 1 * H /*mk*/, m + 2 * H /*mv*/, m + 4 * H /*mg*/,
        part_big, H);
    reduce_kernel<<<dim3(H / 256, 4), 256, 0, stream>>>(part_big, rr, H, NCHUNK, 0, nullptr);

    //    decay LoRA stage 1: dd1 = tanh(mw @ Wd1)   K=4096, N=128
    wmma_gemv_part<<<dim3(1, NCHUNK), wblk, 0, stream>>>(Wd1, m, part_dd, TD);
    reduce_kernel<<<dim3(1, 1), 256, 0, stream>>>(part_dd, dd1, TD, NCHUNK, 1, nullptr);

    // 4) per-head decay/state/attention/InstanceNorm/gate
    head_kernel<<<NH, HS, 0, stream>>>(rr, kk, vv, gp, dd1, Wd2, tdec, faaaa,
                                       st2, lnxw, lnxb, ox, out_s2);

    // 5) o = ox @ Wo;  out_x = x + o
    wmma_gemv_part<<<dim3(H / 128, NCHUNK), wblk, 0, stream>>>(Wo, ox, part_o, H);
    reduce_kernel<<<dim3(H / 256, 1), 256, 0, stream>>>(part_o, out_x, H, NCHUNK, 0, x);
}